// GCNLayer_35527969473088
// MI455X (gfx1250) — compile-verified
//
#include <hip/hip_runtime.h>
#include <stdint.h>

// GCN SpMM scatter: out[rows[e]] += vals[e] * embeds[cols[e]], D=32.
// Memory/atomic bound (102 MFLOP vs ~45MB HBM + L2-resident gather/atomics).
// Wave-autonomous pipeline: each wave double-buffers its own 32-edge metadata
// chunk via gfx1250 async global->LDS loads (ASYNCcnt is per-wave, so no
// workgroup barriers are needed), then does 128B-coalesced embedding gathers
// and global_atomic_add_f32 scatters.

#define D_FEAT 32
#define BLOCK  256            // 8 waves of 32
#define WAVES_PER_WG (BLOCK / 32)
#define CHUNK  32             // edges per wave per pipeline stage

// ---- gfx1250 async global->LDS load (ASYNCcnt path), inline asm ----
// LDS dest from VGPR ldsoff; global src = 64-bit SGPR base + VGPR byte offset
__device__ __forceinline__ void async_ld_b32(uint32_t ldsoff, uint32_t byteoff,
                                             const void* base) {
  asm volatile("global_load_async_to_lds_b32 %0, %1, %2"
               :: "v"(ldsoff), "v"(byteoff), "s"(base)
               : "memory");
}

// Low 32 bits of a flat pointer into __shared__ are the LDS byte offset
// (LDS aperture: addr[63:32]=SHARED_BASE, addr[31:0]=LDS offset).
__device__ __forceinline__ uint32_t lds_off(const void* p) {
  return (uint32_t)(uintptr_t)p;
}

__global__ void gcn_zero_kernel(float* __restrict__ out, int n) {
  int i4 = (blockIdx.x * blockDim.x + threadIdx.x) * 4;
  if (i4 + 3 < n) {
    *(float4*)(out + i4) = make_float4(0.f, 0.f, 0.f, 0.f);
  } else if (i4 < n) {
    for (int k = i4; k < n; ++k) out[k] = 0.f;
  }
}

__global__ __launch_bounds__(BLOCK)
void gcn_spmm_scatter_kernel(const int* __restrict__ rows,
                             const int* __restrict__ cols,
                             const float* __restrict__ vals,
                             const float* __restrict__ embeds,
                             float* __restrict__ out,
                             int nEdges, int nChunks) {
  // Wave-private ping-pong staging: [wave][buf][slot]
  __shared__ int   sRow[WAVES_PER_WG][2][CHUNK];
  __shared__ int   sCol[WAVES_PER_WG][2][CHUNK];
  __shared__ float sVal[WAVES_PER_WG][2][CHUNK];

  const int tid  = threadIdx.x;
  const int wave = tid >> 5;          // 0..7
  const int lane = tid & 31;
  const int eSub = lane >> 3;         // 0..3 : edge within a group of 4
  const int fc   = (lane & 7) * 4;    // 0..28: float4 feature chunk

  const int waveGlobal = blockIdx.x * WAVES_PER_WG + wave;
  const int waveStride = gridDim.x * WAVES_PER_WG;

  // Stage one 32-edge chunk's metadata into this wave's LDS buffer `buf`.
  // Index is clamped (never masked) so the wave always issues exactly 3
  // async loads -> ASYNCcnt bookkeeping below is exact.
  auto stage = [&](int buf, int chunkIdx) {
    int e = chunkIdx * CHUNK + lane;
    if (e >= nEdges) e = nEdges - 1;
    uint32_t boff = (uint32_t)e * 4u;
    async_ld_b32(lds_off(&sRow[wave][buf][lane]), boff, rows);
    async_ld_b32(lds_off(&sCol[wave][buf][lane]), boff, cols);
    async_ld_b32(lds_off(&sVal[wave][buf][lane]), boff, vals);
  };

  int k = waveGlobal;
  if (k >= nChunks) return;

  stage(0, k);
  int buf = 0;

  for (; k < nChunks; k += waveStride) {
    const int kn = k + waveStride;      // wave-uniform
    if (kn < nChunks) {
      // Prefetch next chunk into the other buffer, then drain only the
      // current chunk's 3 loads (async loads complete in order).
      stage(buf ^ 1, kn);
      asm volatile("s_wait_asynccnt 0x3" ::: "memory");
    } else {
      asm volatile("s_wait_asynccnt 0x0" ::: "memory");
    }

    const int base = k * CHUNK;
    #pragma unroll
    for (int it = 0; it < 8; ++it) {
      const int le = it * 4 + eSub;               // edge slot in chunk
      const int ge = base + le;                   // global edge id
      const int   r = sRow[wave][buf][le];        // ds_load broadcast (x8 lanes)
      const int   c = sCol[wave][buf][le];
      const float v = sVal[wave][buf][le];
      if (ge < nEdges) {
        const float4 x = *(const float4*)(embeds + c * D_FEAT + fc); // b128, L2-resident
        float* o = out + r * D_FEAT + fc;
        atomicAdd(o + 0, v * x.x);   // global_atomic_add_f32 (no-return)
        atomicAdd(o + 1, v * x.y);
        atomicAdd(o + 2, v * x.z);
        atomicAdd(o + 3, v * x.w);
      }
    }
    // Values from this buffer are fully consumed above (DScnt waited before
    // use), one iteration before it can be re-staged -> WAR safe, no barrier.
    buf ^= 1;
  }
}

extern "C" void kernel_launch(void* const* d_in, const int* in_sizes, int n_in,
                              void* d_out, int out_size, void* d_ws, size_t ws_size,
                              hipStream_t stream) {
  const int*   rows   = (const int*)d_in[0];
  const int*   cols   = (const int*)d_in[1];
  const float* vals   = (const float*)d_in[2];
  const float* embeds = (const float*)d_in[3];
  float*       out    = (float*)d_out;

  const int nEdges = in_sizes[0];

  // 1) zero the (poisoned) output so atomics accumulate from 0
  {
    int n4 = (out_size + 3) / 4;
    int zb = (n4 + 255) / 256;
    if (zb > 0)
      gcn_zero_kernel<<<zb, 256, 0, stream>>>(out, out_size);
  }

  // 2) edge scatter, wave-autonomous double-buffered async pipeline
  {
    int nChunks = (nEdges + CHUNK - 1) / CHUNK;   // 50000 for E=1.6M
    if (nChunks > 0) {
      int wgNeeded = (nChunks + WAVES_PER_WG - 1) / WAVES_PER_WG;
      int numWG = wgNeeded < 1536 ? wgNeeded : 1536;  // persistent waves, ~4 chunks each
      gcn_spmm_scatter_kernel<<<numWG, BLOCK, 0, stream>>>(
          rows, cols, vals, embeds, out, nEdges, nChunks);
    }
  }
}